// CenterPointPreProcess_83021717832034
// MI455X (gfx1250) — compile-verified
//
#include <hip/hip_runtime.h>
#include <math.h>

// ---------------- static problem config (matches reference) ----------------
#define BATCH   4
#define NPTS    300000
#define NCH     5
#define GXD     512
#define GYD     512
#define GTOTD   (GXD * GYD)          // 262144 voxel bins per batch
#define MAXVX   30000                // max voxels
#define MAXPP   20                   // max points per voxel
#define BV      (BATCH * MAXVX)      // 120000
#define FEATSZ  (NCH * MAXPP * BV)   // 12,000,000 floats
#define CHUNK   256
#define NCHUNK  (GTOTD / CHUNK)      // 1024 chunks per batch
#define TILE    1024                 // points staged per TDM transfer
#define TILE_E  (TILE * NCH)         // 5120 floats = 20 KB LDS per buffer
#define NTB     80                   // persistent blocks per batch (pipelined)

// ---------------- CDNA5 TDM (async tensor) helpers -------------------------
typedef unsigned int u32x4 __attribute__((ext_vector_type(4)));
typedef int          i32x4 __attribute__((ext_vector_type(4)));
typedef int          i32x8 __attribute__((ext_vector_type(8)));

// Issue a 1-D tensor_load_to_lds: `elems` f32 elements starting at `gaddr`
// into LDS byte offset `lds_off`. OOB reads (elems < tile) return zero per ISA.
// Descriptor per CDNA5 ISA D# group0/group1 layout.
__device__ __forceinline__ void tdm_load_tile(unsigned long long gaddr,
                                              unsigned lds_off,
                                              int elems, int tile_elems) {
    u32x4 g0;
    g0.x = 1u;                                           // count=1 (valid), user mode
    g0.y = lds_off;                                      // lds_addr (bytes)
    g0.z = (unsigned)(gaddr & 0xFFFFFFFFull);            // global_addr[31:0]
    g0.w = (unsigned)((gaddr >> 32) & 0x1FFFFFFull)      // global_addr[56:32]
         | (2u << 30);                                   // type=2 ("image")
    i32x8 g1;
    g1[0] = 2 << 16;                                     // data_size=2 -> 4 bytes
    g1[1] = (elems & 0xFFFF) << 16;                      // tensor_dim0[15:0] @ bits63:48
    g1[2] = ((elems >> 16) & 0xFFFF) | (1 << 16);        // tensor_dim0[31:16] | tensor_dim1=1
    g1[3] = (tile_elems & 0xFFFF) << 16;                 // tile_dim0
    g1[4] = 0;                                           // tile_dim1/2 unused (1-D)
    g1[5] = elems;                                       // tensor_dim0_stride (unused for 1-D)
    g1[6] = 0;
    g1[7] = 0;
    i32x4 z4 = {0, 0, 0, 0};
#if __has_include(<hip/amd_detail/amd_gfx1250_TDM.h>)
    i32x8 z8 = {0, 0, 0, 0, 0, 0, 0, 0};
    __builtin_amdgcn_tensor_load_to_lds(g0, g1, z4, z4, z8, 0);   // clang-23 / therock: 6-arg
#else
    __builtin_amdgcn_tensor_load_to_lds(g0, g1, z4, z4, 0);       // ROCm 7.2 clang-22: 5-arg
#endif
}

__device__ __forceinline__ void voxel_coord(float x, float y, float z,
                                            int& cx, int& cy, int& cz) {
    cx = (int)floorf((x + 51.2f) / 0.2f);
    cy = (int)floorf((y + 51.2f) / 0.2f);
    cz = (int)floorf((z + 5.0f) / 8.0f);
}

// ---------------- kernels ---------------------------------------------------

// Zero features, write default coors [b,-1,-1,-1], zero per-voxel counters.
__global__ void k_init(float* __restrict__ out, int* __restrict__ count) {
    const int gid = blockIdx.x * blockDim.x + threadIdx.x;
    const int gsz = gridDim.x * blockDim.x;
    float4 fz = make_float4(0.f, 0.f, 0.f, 0.f);
    float4* f4 = (float4*)out;
    for (int i = gid; i < FEATSZ / 4; i += gsz) f4[i] = fz;
    float4* co = (float4*)(out + FEATSZ);
    for (int i = gid; i < BV; i += gsz)
        co[i] = make_float4((float)(i / MAXVX), -1.0f, -1.0f, -1.0f);
    int4 iz = make_int4(0, 0, 0, 0);
    int4* c4 = (int4*)count;
    for (int i = gid; i < (BATCH * GTOTD) / 4; i += gsz) c4[i] = iz;
}

// Pass 1: histogram points into voxel bins. Persistent blocks walk tiles with
// a double-buffered TDM pipeline (issue next DMA, wait tensorcnt<=1, compute).
__global__ __launch_bounds__(256) void k_count(const float* __restrict__ pts,
                                               int* __restrict__ count) {
    __shared__ float buf[2][TILE_E];   // sole LDS object -> base offset 0
    const int b = blockIdx.y;
    int* cb = count + b * GTOTD;
    const unsigned long long base =
        (unsigned long long)(pts + (unsigned long long)b * NPTS * NCH);
    const int NT = (NPTS + TILE - 1) / TILE;
    const bool dma = (threadIdx.x < 32);   // wave 0 drives the TDM
    int t = blockIdx.x;
    if (t >= NT) return;
    if (dma) {
        int n0 = min(TILE, NPTS - t * TILE);
        tdm_load_tile(base + (unsigned long long)t * TILE * NCH * 4, 0u,
                      n0 * NCH, TILE_E);
    }
    int cur = 0;
    for (; t < NT; t += gridDim.x) {
        const int nt = t + gridDim.x;
        if (dma) {
            if (nt < NT) {   // prefetch next tile into the other buffer
                int n1 = min(TILE, NPTS - nt * TILE);
                tdm_load_tile(base + (unsigned long long)nt * TILE * NCH * 4,
                              (unsigned)((cur ^ 1) * TILE_E * 4), n1 * NCH, TILE_E);
                __builtin_amdgcn_s_wait_tensorcnt(1);   // current tile resident
            } else {
                __builtin_amdgcn_s_wait_tensorcnt(0);
            }
        }
        __syncthreads();                       // buf[cur] visible to all waves
        const int npts_here = min(TILE, NPTS - t * TILE);
        const float* bf = buf[cur];
        for (int k = threadIdx.x; k < npts_here; k += 256) {
            float x = bf[k * NCH + 0];
            float y = bf[k * NCH + 1];
            float z = bf[k * NCH + 2];
            int cx, cy, cz;
            voxel_coord(x, y, z, cx, cy, cz);
            if (cx >= 0 && cx < GXD && cy >= 0 && cy < GYD && cz == 0)
                atomicAdd(&cb[cy * GXD + cx], 1);
        }
        __syncthreads();                       // done reading before buf reuse
        cur ^= 1;
    }
}

// Scan phase A: per-256-bin-chunk occupancy sums.
__global__ __launch_bounds__(256) void k_chunksum(const int* __restrict__ count,
                                                  int* __restrict__ csum) {
    __shared__ int red[CHUNK];
    const int b = blockIdx.y;
    const int bin = blockIdx.x * CHUNK + threadIdx.x;
    red[threadIdx.x] = (count[b * GTOTD + bin] > 0) ? 1 : 0;
    __syncthreads();
    for (int s = CHUNK / 2; s > 0; s >>= 1) {
        if (threadIdx.x < s) red[threadIdx.x] += red[threadIdx.x + s];
        __syncthreads();
    }
    if (threadIdx.x == 0) csum[b * NCHUNK + blockIdx.x] = red[0];
}

// Scan phase B: exclusive scan of the 1024 chunk sums of one batch.
__global__ __launch_bounds__(1024) void k_scan(const int* __restrict__ csum,
                                               int* __restrict__ coff) {
    __shared__ int sh[NCHUNK];
    const int b = blockIdx.x;
    const int t = threadIdx.x;
    const int v = csum[b * NCHUNK + t];
    sh[t] = v;
    __syncthreads();
    for (int d = 1; d < NCHUNK; d <<= 1) {
        int add = (t >= d) ? sh[t - d] : 0;
        __syncthreads();
        sh[t] += add;
        __syncthreads();
    }
    coff[b * NCHUNK + t] = sh[t] - v;   // exclusive prefix
}

// Scan phase C: assign deterministic slot ids (ascending lin id), write coors.
__global__ __launch_bounds__(256) void k_assign(const int* __restrict__ count,
                                                const int* __restrict__ coff,
                                                int* __restrict__ slot,
                                                float* __restrict__ out) {
    __shared__ int sh[CHUNK];
    const int b = blockIdx.y;
    const int t = threadIdx.x;
    const int bin = blockIdx.x * CHUNK + t;
    const int occ = (count[b * GTOTD + bin] > 0) ? 1 : 0;
    sh[t] = occ;
    __syncthreads();
    for (int d = 1; d < CHUNK; d <<= 1) {   // inclusive Hillis-Steele
        int add = (t >= d) ? sh[t - d] : 0;
        __syncthreads();
        sh[t] += add;
        __syncthreads();
    }
    if (occ) {
        int s = coff[b * NCHUNK + blockIdx.x] + sh[t] - 1;
        if (s < MAXVX) {
            slot[b * GTOTD + bin] = s;
            float4* co = (float4*)(out + FEATSZ) + (b * MAXVX + s);
            int cx = bin & (GXD - 1);
            int cy = bin >> 9;
            *co = make_float4((float)b, 0.0f, (float)cy, (float)cx);  // b,z,y,x
        } else {
            slot[b * GTOTD + bin] = -1;
        }
    }
}

// Reset counters so pass 2 can reuse them as per-voxel rank counters.
__global__ void k_rezero(int* __restrict__ count) {
    const int gid = blockIdx.x * blockDim.x + threadIdx.x;
    const int gsz = gridDim.x * blockDim.x;
    int4 iz = make_int4(0, 0, 0, 0);
    int4* c4 = (int4*)count;
    for (int i = gid; i < (BATCH * GTOTD) / 4; i += gsz) c4[i] = iz;
}

// Pass 2: TDM-pipelined point stream; look up slot, take a rank, write the 5
// channels (x,y,z normalized) directly into the transposed [C, MAXP, B*MAXV] layout.
__global__ __launch_bounds__(256) void k_scatter(const float* __restrict__ pts,
                                                 const float* __restrict__ nr,
                                                 const int* __restrict__ slot,
                                                 int* __restrict__ count,
                                                 float* __restrict__ out) {
    __shared__ float buf[2][TILE_E];
    const int b = blockIdx.y;
    const int* sl = slot + b * GTOTD;
    int* cb = count + b * GTOTD;
    const unsigned long long base =
        (unsigned long long)(pts + (unsigned long long)b * NPTS * NCH);
    const int NT = (NPTS + TILE - 1) / TILE;
    const bool dma = (threadIdx.x < 32);
    // normalization: start=nr[0:3], scale=nr[3:6]-nr[0:3]
    const float s0 = nr[0], s1 = nr[1], s2 = nr[2];
    const float r0 = nr[3] - s0, r1 = nr[4] - s1, r2 = nr[5] - s2;
    int t = blockIdx.x;
    if (t >= NT) return;
    if (dma) {
        int n0 = min(TILE, NPTS - t * TILE);
        tdm_load_tile(base + (unsigned long long)t * TILE * NCH * 4, 0u,
                      n0 * NCH, TILE_E);
    }
    int cur = 0;
    for (; t < NT; t += gridDim.x) {
        const int nt = t + gridDim.x;
        if (dma) {
            if (nt < NT) {
                int n1 = min(TILE, NPTS - nt * TILE);
                tdm_load_tile(base + (unsigned long long)nt * TILE * NCH * 4,
                              (unsigned)((cur ^ 1) * TILE_E * 4), n1 * NCH, TILE_E);
                __builtin_amdgcn_s_wait_tensorcnt(1);
            } else {
                __builtin_amdgcn_s_wait_tensorcnt(0);
            }
        }
        __syncthreads();
        const int npts_here = min(TILE, NPTS - t * TILE);
        const float* bf = buf[cur];
        for (int k = threadIdx.x; k < npts_here; k += 256) {
            float x  = bf[k * NCH + 0];
            float y  = bf[k * NCH + 1];
            float z  = bf[k * NCH + 2];
            float f3 = bf[k * NCH + 3];
            float f4 = bf[k * NCH + 4];
            int cx, cy, cz;
            voxel_coord(x, y, z, cx, cy, cz);
            if (cx >= 0 && cx < GXD && cy >= 0 && cy < GYD && cz == 0) {
                int lin = cy * GXD + cx;
                int s = sl[lin];
                if (s >= 0) {
                    int r = atomicAdd(&cb[lin], 1);
                    if (r < MAXPP) {
                        size_t v = (size_t)(b * MAXVX + s);
                        out[(0 * MAXPP + r) * (size_t)BV + v] = (x - s0) / r0;
                        out[(1 * MAXPP + r) * (size_t)BV + v] = (y - s1) / r1;
                        out[(2 * MAXPP + r) * (size_t)BV + v] = (z - s2) / r2;
                        out[(3 * MAXPP + r) * (size_t)BV + v] = f3;
                        out[(4 * MAXPP + r) * (size_t)BV + v] = f4;
                    }
                }
            }
        }
        __syncthreads();
        cur ^= 1;
    }
}

// ---------------- launch ----------------------------------------------------
extern "C" void kernel_launch(void* const* d_in, const int* in_sizes, int n_in,
                              void* d_out, int out_size, void* d_ws, size_t ws_size,
                              hipStream_t stream) {
    const float* pts = (const float*)d_in[0];   // [B, N, C] f32
    const float* nr  = (const float*)d_in[1];   // [6] f32 norm_range
    float* out = (float*)d_out;                 // 12,000,000 feat + 480,000 coors

    // workspace layout (~8.4 MB)
    int* count = (int*)d_ws;                    // B*GTOT
    int* slot  = count + BATCH * GTOTD;         // B*GTOT
    int* csum  = slot + BATCH * GTOTD;          // B*NCHUNK
    int* coff  = csum + BATCH * NCHUNK;         // B*NCHUNK

    k_init<<<2048, 256, 0, stream>>>(out, count);
    k_count<<<dim3(NTB, BATCH), 256, 0, stream>>>(pts, count);
    k_chunksum<<<dim3(NCHUNK, BATCH), CHUNK, 0, stream>>>(count, csum);
    k_scan<<<BATCH, NCHUNK, 0, stream>>>(csum, coff);
    k_assign<<<dim3(NCHUNK, BATCH), CHUNK, 0, stream>>>(count, coff, slot, out);
    k_rezero<<<512, 256, 0, stream>>>(count);
    k_scatter<<<dim3(NTB, BATCH), 256, 0, stream>>>(pts, nr, slot, count, out);
}